// GCN_1786706395639
// MI455X (gfx1250) — compile-verified
//
#include <hip/hip_runtime.h>

typedef __attribute__((ext_vector_type(2))) float v2f;
typedef __attribute__((ext_vector_type(8))) float v8f;

#define GNN_N_NODES 100000

// Fused dual GEMM: given IN [N,K], computes
//   yrel [N,HO] = act(IN) @ Wrel.T          (edge-message features, pre-aggregation)
//   hout [N,HO] = act(IN) @ Wroot.T + bias  (accumulator init for edge scatter)
// act = leaky_relu(0.01) if LRELU_IN else identity.
// One wave computes a 16-row tile using V_WMMA_F32_16X16X4_F32.
template<int K, int HO, bool LRELU_IN>
__global__ __launch_bounds__(256)
void gemm_dual(const float* __restrict__ in,
               const float* __restrict__ Wrel,   // [HO, K] row-major
               const float* __restrict__ Wroot,  // [HO, K] row-major
               const float* __restrict__ bias,   // [HO]
               float* __restrict__ yrel,         // [N, HO]
               float* __restrict__ hout)         // [N, HO]
{
  const int lane = threadIdx.x & 31;
  const int wave = (int)((blockIdx.x * blockDim.x + threadIdx.x) >> 5);
  const int row0 = wave * 16;
  if (row0 >= GNN_N_NODES) return;   // wave-uniform: EXEC stays all-ones for WMMA

  const int r  = lane & 15;          // A: row in tile; B/C/D: column
  const int kh = (lane >> 4) * 2;    // A/B: k-offset within 4-group
  const int mh = (lane >> 4) * 8;    // C/D: row offset

  constexpr int NT = (HO + 15) / 16; // 16-wide column tiles
  v8f acc_rel[NT];
  v8f acc_root[NT];
  v8f zero = {};
#pragma unroll
  for (int t = 0; t < NT; ++t) { acc_rel[t] = zero; acc_root[t] = zero; }

  const float* inrow = in + (size_t)(row0 + r) * K;

  for (int k0 = 0; k0 < K; k0 += 4) {
    // A 16x4 f32: lane r holds rows, VGPR0/1 hold k = k0+kh, k0+kh+1
    float a0 = inrow[k0 + kh];
    float a1 = inrow[k0 + kh + 1];
    if (LRELU_IN) {
      a0 = (a0 > 0.f) ? a0 : 0.01f * a0;
      a1 = (a1 > 0.f) ? a1 : 0.01f * a1;
    }
    v2f a; a.x = a0; a.y = a1;
#pragma unroll
    for (int t = 0; t < NT; ++t) {
      const int col = t * 16 + r;
      v2f brel, broot;
      if (col < HO) {
        // B 4x16 f32: B[k][n] = W[n][k]; lane = n, k = k0+kh (+1)
        brel.x  = Wrel [col * K + k0 + kh];
        brel.y  = Wrel [col * K + k0 + kh + 1];
        broot.x = Wroot[col * K + k0 + kh];
        broot.y = Wroot[col * K + k0 + kh + 1];
      } else {
        brel.x = 0.f; brel.y = 0.f; broot.x = 0.f; broot.y = 0.f;
      }
      acc_rel[t]  = __builtin_amdgcn_wmma_f32_16x16x4_f32(
          false, a, false, brel,  (short)0, acc_rel[t],  false, false);
      acc_root[t] = __builtin_amdgcn_wmma_f32_16x16x4_f32(
          false, a, false, broot, (short)0, acc_root[t], false, false);
    }
  }

  // C/D layout: element v -> row = row0 + mh + v, col = t*16 + r
#pragma unroll
  for (int t = 0; t < NT; ++t) {
    const int col = t * 16 + r;
    if (col < HO) {
      const float b = bias[col];
#pragma unroll
      for (int v = 0; v < 8; ++v) {
        const size_t off = (size_t)(row0 + mh + v) * HO + col;
        yrel[off] = acc_rel[t][v];
        hout[off] = acc_root[t][v] + b;
      }
    }
  }
}

// Edge scatter: h[dst, f] += ew * y[src, f]. F lanes per edge (F = 32 or 8),
// coalesced gathers/atomics; feature buffers (<=12.8 MB) live in L2.
template<int F>
__global__ __launch_bounds__(256)
void edge_scatter(const int* __restrict__ ei,   // [2, E]: src row then dst row
                  const float* __restrict__ ew, // [E]
                  const float* __restrict__ y,  // [N, F]
                  float* __restrict__ h,        // [N, F]
                  int n_edges)
{
  const long long gid = (long long)blockIdx.x * blockDim.x + threadIdx.x;
  const int e = (int)(gid >> (F == 32 ? 5 : 3));
  if (e >= n_edges) return;
  const int f = (int)(gid & (F - 1));
  const int src = ei[e];
  const int dst = ei[n_edges + e];
  const float w = ew[e];
  const float val = w * y[(size_t)src * F + f];
  unsafeAtomicAdd(&h[(size_t)dst * F + f], val);  // -> global_atomic_add_f32
}

extern "C" void kernel_launch(void* const* d_in, const int* in_sizes, int n_in,
                              void* d_out, int out_size, void* d_ws, size_t ws_size,
                              hipStream_t stream) {
  const float* x       = (const float*)d_in[0];
  const int*   ei      = (const int*)  d_in[1];
  const float* ew      = (const float*)d_in[2];
  const float* W1_rel  = (const float*)d_in[3];
  const float* b1      = (const float*)d_in[4];
  const float* W1_root = (const float*)d_in[5];
  const float* W2_rel  = (const float*)d_in[6];
  const float* b2      = (const float*)d_in[7];
  const float* W2_root = (const float*)d_in[8];
  const float* W3_rel  = (const float*)d_in[9];
  const float* b3      = (const float*)d_in[10];
  const float* W3_root = (const float*)d_in[11];
  float* out = (float*)d_out;

  const int E = in_sizes[2];

  float* ws   = (float*)d_ws;
  float* ybuf = ws;                           // reused scratch: [N,32] (or [N,8])
  float* h1   = ws + (size_t)GNN_N_NODES * 32;
  float* h2   = h1 + (size_t)GNN_N_NODES * 32;

  const int tiles   = (GNN_N_NODES + 15) / 16;
  const int gblocks = (tiles + 7) / 8;        // 8 waves (tiles) per 256-thread block

  const int eb32 = (int)(((long long)E * 32 + 255) / 256);
  const int eb8  = (int)(((long long)E * 8  + 255) / 256);

  // Layer 1: project 128->32 first, aggregate in 32-dim
  gemm_dual<128, 32, false><<<gblocks, 256, 0, stream>>>(x, W1_rel, W1_root, b1, ybuf, h1);
  edge_scatter<32><<<eb32, 256, 0, stream>>>(ei, ew, ybuf, h1, E);

  // Layer 2 (leaky-ReLU applied on load of h1)
  gemm_dual<32, 32, true><<<gblocks, 256, 0, stream>>>(h1, W2_rel, W2_root, b2, ybuf, h2);
  edge_scatter<32><<<eb32, 256, 0, stream>>>(ei, ew, ybuf, h2, E);

  // Layer 3: project 32->8 first, aggregate in 8-dim directly into d_out
  gemm_dual<32, 8, true><<<gblocks, 256, 0, stream>>>(h2, W3_rel, W3_root, b3, ybuf, out);
  edge_scatter<8><<<eb8, 256, 0, stream>>>(ei, ew, ybuf, out, E);
}